// WassOrderDistanceGromov_49787260895824
// MI455X (gfx1250) — compile-verified
//
#include <hip/hip_runtime.h>
#include <math.h>

// ---------------------------------------------------------------------------
// WassOrderDistanceGromov for MI455X (gfx1250).
// fp32 matrix ops via V_WMMA_F32_16X16X4_F32 (wave32):
//   A 16x4 : lane(l<16) v0/v1 = A[M=l, k-slot0/1]; lanes 16-31 = k-slot2/3
//   B 4x16 : same pattern with N=lane
//   C/D    : vgpr g: lanes0-15 -> (M=g, N=l), lanes16-31 -> (M=g+8, N=l)
// K-slot assignment inside one WMMA is free (sum over k commutes).
// Hot GEMMs stage operands into LDS with GLOBAL_LOAD_ASYNC_TO_LDS_B128
// (ASYNCcnt), double-buffered so DMA of slab k+2 overlaps WMMA on slab k+1.
// s_wait_asynccnt 6 exploits in-order async-load completion: waiting down to
// the 6 newest per-wave copies guarantees the current slab has landed.
// ---------------------------------------------------------------------------

typedef float v2f __attribute__((ext_vector_type(2)));
typedef float v4f __attribute__((ext_vector_type(4)));
typedef float v8f __attribute__((ext_vector_type(8)));

#define TPB 256
constexpr int BS = 8;
constexpr int NN = 1024;   // n == m
constexpr int DD = 512;
constexpr int DOT_BLK = 64;
constexpr float EPSV = 1e-12f;
constexpr float BETA_T = 0.1f;   // threshold beta

// TN GEMM tiling
constexpr int KB  = 32;          // K slab staged in LDS
constexpr int AST = 136;         // A slab LDS row stride (128 + 8 pad floats)
constexpr int BST = 72;          // B slab LDS row stride (64 + 8 pad floats)

#define WMMA_F32(a, b, c)                                                     \
  __builtin_amdgcn_wmma_f32_16x16x4_f32(false, (a), false, (b), (short)0,     \
                                        (c), false, false)

// ------------------------------ elementwise --------------------------------

__global__ void fill_k(float* __restrict__ p, unsigned total, float v) {
  unsigned i = blockIdx.x * TPB + threadIdx.x;
  if (i < total) p[i] = v;
}

__global__ void exp_k(const float* __restrict__ c, float* __restrict__ a,
                      unsigned total, float invb) {
  unsigned i = blockIdx.x * TPB + threadIdx.x;
  if (i < total) a[i] = expf(-c[i] * invb);
}

__global__ void thresh_k(float* __restrict__ x, unsigned total,
                         const float* __restrict__ mm) {
  unsigned i = blockIdx.x * TPB + threadIdx.x;
  if (i >= total) return;
  float mn = mm[0], mx = mm[1];
  float t = mn + BETA_T * (mx - mn);
  float v = x[i] - t;
  x[i] = v > 0.f ? v : 0.f;
}

// T[b,i,j] = delta[b,i] * (A*T)[b,i,j] * sigma[b,j]   (in-place, elementwise)
__global__ void updT_k(const float* __restrict__ A, float* __restrict__ T,
                       const float* __restrict__ delta,
                       const float* __restrict__ sigma) {
  unsigned idx = blockIdx.x * TPB + threadIdx.x;          // exact: BS*NN*NN
  unsigned j   = idx & (NN - 1);
  unsigned row = idx >> 10;                               // b*NN + i
  unsigned b   = row >> 10;
  T[idx] = delta[row] * A[idx] * T[idx] * sigma[(b << 10) + j];
}

// ------------------------------ reductions ---------------------------------

__device__ __forceinline__ float block_reduce_sum(float s, float* red) {
  int tid = threadIdx.x;
  red[tid] = s; __syncthreads();
  for (int st = TPB / 2; st > 0; st >>= 1) {
    if (tid < st) red[tid] += red[tid + st];
    __syncthreads();
  }
  return red[0];
}

__global__ void normalize_rows_k(const float* __restrict__ in,
                                 float* __restrict__ out) {
  __shared__ float red[TPB];
  size_t row = blockIdx.x;
  const float* r = in + row * DD;
  float* o = out + row * DD;
  int tid = threadIdx.x;
  float s = 0.f;
  for (int d = tid; d < DD; d += TPB) { float v = r[d]; s += v * v; }
  float tot = block_reduce_sum(s, red);
  float inv = 1.f / (sqrtf(tot) + EPSV);
  for (int d = tid; d < DD; d += TPB) o[d] = r[d] * inv;
}

// outv[row] = scale * sum_j in[row,j]^2      (row length NN)
__global__ void rowsq_k(const float* __restrict__ in, float* __restrict__ outv,
                        float scale) {
  __shared__ float red[TPB];
  size_t row = blockIdx.x;
  const float* r = in + row * (size_t)NN;
  float s = 0.f;
  for (int j = threadIdx.x; j < NN; j += TPB) { float v = r[j]; s += v * v; }
  float tot = block_reduce_sum(s, red);
  if (threadIdx.x == 0) outv[row] = scale * tot;
}

__global__ void minmax_part_k(const float* __restrict__ x, unsigned total,
                              float* __restrict__ part) {
  __shared__ float smn[TPB], smx[TPB];
  int tid = threadIdx.x;
  float mn = 3.4e38f, mx = -3.4e38f;
  for (unsigned i = blockIdx.x * TPB + tid; i < total; i += gridDim.x * TPB) {
    float v = x[i]; mn = fminf(mn, v); mx = fmaxf(mx, v);
  }
  smn[tid] = mn; smx[tid] = mx; __syncthreads();
  for (int st = TPB / 2; st > 0; st >>= 1) {
    if (tid < st) {
      smn[tid] = fminf(smn[tid], smn[tid + st]);
      smx[tid] = fmaxf(smx[tid], smx[tid + st]);
    }
    __syncthreads();
  }
  if (tid == 0) { part[blockIdx.x * 2] = smn[0]; part[blockIdx.x * 2 + 1] = smx[0]; }
}

__global__ void minmax_final_k(const float* __restrict__ part, int nparts,
                               float* __restrict__ mm) {
  __shared__ float smn[TPB], smx[TPB];
  int tid = threadIdx.x;
  float mn = 3.4e38f, mx = -3.4e38f;
  for (int i = tid; i < nparts; i += TPB) {
    mn = fminf(mn, part[i * 2]);
    mx = fmaxf(mx, part[i * 2 + 1]);
  }
  smn[tid] = mn; smx[tid] = mx; __syncthreads();
  for (int st = TPB / 2; st > 0; st >>= 1) {
    if (tid < st) {
      smn[tid] = fminf(smn[tid], smn[tid + st]);
      smx[tid] = fmaxf(smx[tid], smx[tid + st]);
    }
    __syncthreads();
  }
  if (tid == 0) { mm[0] = smn[0]; mm[1] = smx[0]; }
}

// delta[b,i] = 1 / (n * sum_j A[b,i,j]*T[b,i,j]*sigma[b,j])
__global__ void row_delta_k(const float* __restrict__ A,
                            const float* __restrict__ T,
                            const float* __restrict__ sigma,
                            float* __restrict__ delta) {
  __shared__ float red[TPB];
  int b = blockIdx.y, i = blockIdx.x;
  size_t row = ((size_t)b * NN + i) * NN;
  const float* sb = sigma + (size_t)b * NN;
  float s = 0.f;
  for (int j = threadIdx.x; j < NN; j += TPB)
    s += A[row + j] * T[row + j] * sb[j];
  float tot = block_reduce_sum(s, red);
  if (threadIdx.x == 0) delta[(size_t)b * NN + i] = 1.f / ((float)NN * tot);
}

// sigma[b,j] = 1 / (m * sum_i A[b,i,j]*T[b,i,j]*delta[b,i])   (coalesced in j)
__global__ void col_sigma_k(const float* __restrict__ A,
                            const float* __restrict__ T,
                            const float* __restrict__ delta,
                            float* __restrict__ sigma) {
  int b = blockIdx.y;
  int j = blockIdx.x * TPB + threadIdx.x;
  const float* db = delta + (size_t)b * NN;
  size_t base = (size_t)b * NN * NN + j;
  float s = 0.f;
  for (int i = 0; i < NN; ++i)
    s += A[base + (size_t)i * NN] * T[base + (size_t)i * NN] * db[i];
  sigma[(size_t)b * NN + j] = 1.f / ((float)NN * s);
}

// part[b, blk] = partial of sum(a*b) over batch slice
__global__ void dot_part_k(const float* __restrict__ a,
                           const float* __restrict__ bb,
                           float* __restrict__ part) {
  __shared__ float red[TPB];
  int b = blockIdx.y;
  size_t base = (size_t)b * NN * NN;
  float s = 0.f;
  for (unsigned i = blockIdx.x * TPB + threadIdx.x; i < (unsigned)NN * NN;
       i += DOT_BLK * TPB)
    s += a[base + i] * bb[base + i];
  float tot = block_reduce_sum(s, red);
  if (threadIdx.x == 0) part[b * DOT_BLK + blockIdx.x] = tot;
}

__global__ void final_k(const float* __restrict__ wdp,
                        const float* __restrict__ gwdp, int count,
                        float* __restrict__ out) {
  __shared__ float red[TPB];
  float s = 0.f;
  for (int i = threadIdx.x; i < count; i += TPB) s += wdp[i] + gwdp[i];
  float tot = block_reduce_sum(s, red);
  if (threadIdx.x == 0) out[0] = tot / (float)BS;
}

// ------------------------------ WMMA GEMMs ---------------------------------
// NT: C[b,i,j] = epi( sum_k A[b,i,k]*B[b,j,k] )   A:[bs,M,K] B:[bs,N,K]
// Each wave computes a 16x64 strip: one A fragment feeds 4 WMMAs.
// EPI: 0 = acc, 1 = 1 - acc
template <int EPI>
__global__ __launch_bounds__(TPB) void gemm_nt_k(const float* __restrict__ A,
                                                 const float* __restrict__ B,
                                                 float* __restrict__ C, int M,
                                                 int N, int K) {
  int b = blockIdx.z;
  int wave = threadIdx.x >> 5, lane = threadIdx.x & 31;
  int lrow = lane & 15, half = lane >> 4;
  int tiles64 = N >> 6;
  int tile = blockIdx.x * 8 + wave;
  int tI = tile / tiles64, tJ = tile - tI * tiles64;
  const float* Ar = A + (size_t)b * M * K + (size_t)(tI * 16 + lrow) * K + 4 * half;
  const float* Br = B + (size_t)b * N * K + (size_t)(tJ * 64 + lrow) * K + 4 * half;
  size_t bstep = (size_t)16 * K;   // 16 rows of B per sub-tile
  v8f acc[4];
#pragma unroll
  for (int q = 0; q < 4; ++q)
    acc[q] = (v8f){0.f, 0.f, 0.f, 0.f, 0.f, 0.f, 0.f, 0.f};
#pragma unroll 2
  for (int kk = 0; kk < K; kk += 8) {
    v4f a4 = *(const v4f*)(Ar + kk);
    v2f a0 = {a4.x, a4.y};   // k-slots {kk,kk+1 | kk+4,kk+5}
    v2f a1 = {a4.z, a4.w};   // k-slots {kk+2,kk+3 | kk+6,kk+7}
#pragma unroll
    for (int q = 0; q < 4; ++q) {
      v4f b4 = *(const v4f*)(Br + q * bstep + kk);
      v2f b0 = {b4.x, b4.y};
      v2f b1 = {b4.z, b4.w};
      acc[q] = WMMA_F32(a0, b0, acc[q]);
      acc[q] = WMMA_F32(a1, b1, acc[q]);
    }
  }
  float* Cr = C + (size_t)b * M * N;
#pragma unroll
  for (int q = 0; q < 4; ++q) {
    int col = tJ * 64 + q * 16 + lrow;
#pragma unroll
    for (int g = 0; g < 8; ++g) {
      int r = tI * 16 + half * 8 + g;
      float v = acc[q][g];
      if (EPI == 1) v = 1.f - v;
      Cr[(size_t)r * N + col] = v;
    }
  }
}

// Issue one K-slab's async copies (6 per thread => 6 per-wave async ops).
__device__ __forceinline__ void issue_slab(const float* __restrict__ Ab,
                                           const float* __restrict__ Bb,
                                           int kb, int M, int N, unsigned aoff,
                                           unsigned boff, int t) {
  // A slab: KB x 128 floats, contiguous per k-row -> 1024 16B lane-copies
#pragma unroll
  for (int r = 0; r < 4; ++r) {
    int ta = t + r * TPB;
    int row = ta >> 5, ch = ta & 31;
    const float* g = Ab + (size_t)(kb + row) * M + ch * 4;
    unsigned l = aoff + (unsigned)(row * AST + ch * 4) * 4u;
    asm volatile("global_load_async_to_lds_b128 %0, %1, off"
                 :: "v"(l), "v"(g) : "memory");
  }
  // B slab: KB x 64 floats -> 512 16B lane-copies
#pragma unroll
  for (int r = 0; r < 2; ++r) {
    int tb = t + r * TPB;
    int row = tb >> 4, ch = tb & 15;
    const float* g = Bb + (size_t)(kb + row) * N + ch * 4;
    unsigned l = boff + (unsigned)(row * BST + ch * 4) * 4u;
    asm volatile("global_load_async_to_lds_b128 %0, %1, off"
                 :: "v"(l), "v"(g) : "memory");
  }
}

// TN: out[i,j] = epi( sum_k A[b,k,i]*B[b,k,j] )   A:[bs,K,M] B:[bs,K,N]
// Block = 8 waves = 128x64 tile; double-buffered async LDS staging.
// EPI: 0 = acc, 2 = rowv[b,i] + colv[b,j] - 2*acc
// STORE_T: store out transposed ([bs,N,M])
template <int EPI, int STORE_T>
__global__ __launch_bounds__(TPB) void gemm_tn_k(
    const float* __restrict__ A, const float* __restrict__ B,
    float* __restrict__ C, int M, int N, int K, const float* __restrict__ rowv,
    const float* __restrict__ colv) {
  __shared__ float Asl[2][KB * AST];   // [buf][k][i-local], padded rows
  __shared__ float Bsl[2][KB * BST];   // [buf][k][j-local], padded rows
  int b = blockIdx.z;
  int nb = N >> 6;
  int bi = blockIdx.x / nb, bj = blockIdx.x - bi * nb;
  int iBase = bi * 128, jBase = bj * 64;
  int t = threadIdx.x;
  int wave = t >> 5, lane = t & 31, lrow = lane & 15, half = lane >> 4;

  const float* Ab = A + (size_t)b * K * M + iBase;
  const float* Bb = B + (size_t)b * K * N + jBase;
  unsigned aoff[2] = {(unsigned)(size_t)&Asl[0][0], (unsigned)(size_t)&Asl[1][0]};
  unsigned boff[2] = {(unsigned)(size_t)&Bsl[0][0], (unsigned)(size_t)&Bsl[1][0]};

  v8f acc[4];
#pragma unroll
  for (int q = 0; q < 4; ++q)
    acc[q] = (v8f){0.f, 0.f, 0.f, 0.f, 0.f, 0.f, 0.f, 0.f};

  // Prologue: prefetch slabs 0 and 1 (12 outstanding per-wave async ops).
  issue_slab(Ab, Bb, 0, M, N, aoff[0], boff[0], t);
  issue_slab(Ab, Bb, KB, M, N, aoff[1], boff[1], t);

  int cur = 0;
  for (int kb = 0; kb < K; kb += KB, cur ^= 1) {
    // In-order completion: allowing the 6 newest copies to remain outstanding
    // guarantees the 6 older ones (current slab) have landed.
    if (kb + KB < K)
      asm volatile("s_wait_asynccnt 0x6" ::: "memory");
    else
      asm volatile("s_wait_asynccnt 0x0" ::: "memory");
    __syncthreads();                     // slab visible to all waves
    const float* As = &Asl[cur][0];
    const float* Bs = &Bsl[cur][0];
#pragma unroll
    for (int kk = 0; kk < KB; kk += 4) {
      int k0 = kk + 2 * half;            // k-slots {kk..kk+3}
      int ai = k0 * AST + wave * 16 + lrow;
      v2f a = {As[ai], As[ai + AST]};
      int bx = k0 * BST + lrow;
#pragma unroll
      for (int q = 0; q < 4; ++q) {
        v2f bb = {Bs[bx + q * 16], Bs[bx + BST + q * 16]};
        acc[q] = WMMA_F32(a, bb, acc[q]);
      }
    }
    __syncthreads();                     // all waves done reading this buffer
    if (kb + 2 * KB < K)                 // prefetch slab kb+2 into freed buffer
      issue_slab(Ab, Bb, kb + 2 * KB, M, N, aoff[cur], boff[cur], t);
  }

#pragma unroll
  for (int q = 0; q < 4; ++q) {
    int col = jBase + q * 16 + lrow;
#pragma unroll
    for (int g = 0; g < 8; ++g) {
      int r = iBase + wave * 16 + half * 8 + g;
      float v = acc[q][g];
      if (EPI == 2)
        v = rowv[(size_t)b * M + r] + colv[(size_t)b * N + col] - 2.f * v;
      if (STORE_T)
        C[(size_t)b * N * M + (size_t)col * M + r] = v;
      else
        C[(size_t)b * M * N + (size_t)r * N + col] = v;
    }
  }
}

// ------------------------------ host orchestration -------------------------

static void run_thresh(float* buf, float* part, float* mm, hipStream_t st) {
  unsigned total = (unsigned)BS * NN * NN;
  minmax_part_k<<<1024, TPB, 0, st>>>(buf, total, part);
  minmax_final_k<<<1, TPB, 0, st>>>(part, 1024, mm);
  thresh_k<<<total / TPB, TPB, 0, st>>>(buf, total, mm);
}

static void run_ipot(const float* C, float* Aew, float* T, float* delta,
                     float* sigma, float beta, int iters, hipStream_t st) {
  unsigned total = (unsigned)BS * NN * NN;
  unsigned eb = total / TPB;
  exp_k<<<eb, TPB, 0, st>>>(C, Aew, total, 1.f / beta);
  fill_k<<<eb, TPB, 0, st>>>(T, total, 1.f);
  fill_k<<<(BS * NN + TPB - 1) / TPB, TPB, 0, st>>>(sigma, BS * NN, 1.f / NN);
  for (int it = 0; it < iters; ++it) {
    row_delta_k<<<dim3(NN, BS), TPB, 0, st>>>(Aew, T, sigma, delta);
    col_sigma_k<<<dim3(NN / TPB, BS), TPB, 0, st>>>(Aew, T, delta, sigma);
    updT_k<<<eb, TPB, 0, st>>>(Aew, T, delta, sigma);
  }
}

static void gw_cgamma(const float* Cs, const float* Ct, const float* gamma,
                      float* tmpT, float* Cgam, const float* rowv,
                      const float* colv, hipStream_t st) {
  dim3 grid((NN / 128) * (NN / 64), 1, BS);
  // tmpT = (Cs @ gamma)^T   using Cs symmetry: TN form, coalesced slabs
  gemm_tn_k<0, 1><<<grid, TPB, 0, st>>>(Cs, gamma, tmpT, NN, NN, NN, nullptr,
                                        nullptr);
  // Cgam = rowv + colv - 2 * (tmp @ Ct)   (Ct symmetric => Ct == Ct^T)
  gemm_tn_k<2, 0><<<grid, TPB, 0, st>>>(tmpT, Ct, Cgam, NN, NN, NN, rowv, colv);
}

extern "C" void kernel_launch(void* const* d_in, const int* in_sizes, int n_in,
                              void* d_out, int out_size, void* d_ws,
                              size_t ws_size, hipStream_t stream) {
  (void)in_sizes; (void)n_in; (void)out_size; (void)ws_size;
  const float* X = (const float*)d_in[0];   // [8,1024,512]
  const float* Y = (const float*)d_in[1];   // [8,1024,512]
  float* out = (float*)d_out;               // scalar

  float* w = (float*)d_ws;
  const size_t P = (size_t)BS * NN * NN;    // 8M floats
  const size_t Pn = (size_t)BS * NN * DD;   // 4M floats
  float* Xn = w;                 // phase-1 only
  float* Yn = w + Pn;            // phase-1 only
  float* gamma = w;              // aliases Xn|Yn (exactly P floats), phase-2
  float* Cxy = w + 2 * Pn;       // later reused as C_gamma
  float* Cs = Cxy + P;
  float* Ct = Cs + P;
  float* Aew = Ct + P;           // exp(-C/beta) scratch
  float* Tw = Aew + P;           // first-IPOT plan; later tmpT
  float* delta = Tw + P;         // BS*NN
  float* sigma = delta + BS * NN;
  float* rowv = sigma + BS * NN;
  float* colv = rowv + BS * NN;
  float* mm = colv + BS * NN;    // 2
  float* part = mm + 2;          // 2048
  float* wdp = part + 2048;      // BS*DOT_BLK
  float* gwdp = wdp + BS * DOT_BLK;

  // 1) l2-normalize rows of X, Y (== normalize along D of X^T)
  normalize_rows_k<<<BS * NN, TPB, 0, stream>>>(X, Xn);
  normalize_rows_k<<<BS * NN, TPB, 0, stream>>>(Y, Yn);

  // 2) cosine-distance matrices via f32 WMMA (NT GEMM, epilogue 1-acc)
  dim3 gNT((NN / 16) * (NN / 64) / 8, 1, BS);
  gemm_nt_k<1><<<gNT, TPB, 0, stream>>>(Xn, Yn, Cxy, NN, NN, DD);
  gemm_nt_k<1><<<gNT, TPB, 0, stream>>>(Xn, Xn, Cs, NN, NN, DD);
  gemm_nt_k<1><<<gNT, TPB, 0, stream>>>(Yn, Yn, Ct, NN, NN, DD);

  // 3) global min/max threshold-relu (per tensor)
  run_thresh(Cxy, part, mm, stream);
  run_thresh(Cs, part, mm, stream);
  run_thresh(Ct, part, mm, stream);

  // 4) wd = <Cxy, IPOT(Cxy, 0.5, 30)>
  run_ipot(Cxy, Aew, Tw, delta, sigma, 0.5f, 30, stream);
  dot_part_k<<<dim3(DOT_BLK, BS), TPB, 0, stream>>>(Cxy, Tw, wdp);

  // 5) Gromov-Wasserstein
  rowsq_k<<<BS * NN, TPB, 0, stream>>>(Cs, rowv, 1.f / NN);  // (Cs^2) @ p
  rowsq_k<<<BS * NN, TPB, 0, stream>>>(Ct, colv, 1.f / NN);  // (Ct^2) @ q
  fill_k<<<P / TPB, TPB, 0, stream>>>(gamma, (unsigned)P,
                                      1.f / ((float)NN * (float)NN));
  for (int it = 0; it < 5; ++it) {
    gw_cgamma(Cs, Ct, gamma, Tw, Cxy, rowv, colv, stream);   // Cxy := C_gamma
    run_ipot(Cxy, Aew, gamma, delta, sigma, 0.1f, 20, stream);
  }
  gw_cgamma(Cs, Ct, gamma, Tw, Cxy, rowv, colv, stream);
  dot_part_k<<<dim3(DOT_BLK, BS), TPB, 0, stream>>>(Cxy, gamma, gwdp);

  // 6) out = mean(gwd) + mean(wd)
  final_k<<<1, TPB, 0, stream>>>(wdp, gwdp, BS * DOT_BLK, out);
}